// HydraTransform_32066225832351
// MI455X (gfx1250) — compile-verified
//
#include <hip/hip_runtime.h>

typedef __attribute__((ext_vector_type(2))) float v2f;
typedef __attribute__((ext_vector_type(8))) float v8f;

#define T_FULL 75000
#define CH_IN  17
#define NGROUP 32
#define NKER   8
#define NTAP   9
#define NUNIT  28                 // 14 dilations x {x, diff}
#define SG_FLOATS 75040           // g sequence (padded)
#define ZSLOT     75008           // dedicated always-zero LDS slot (>= Teff)
#define SW_OFF    SG_FLOATS       // 72 weights
#define SACC_OFF  (SG_FLOATS + 72)
#define SMEM_FLOATS (SACC_OFF + 16)

__global__ __launch_bounds__(1024)
void hydra_unit_kernel(const float* __restrict__ X,
                       const float* __restrict__ W,
                       const int*   __restrict__ IDX,
                       float* __restrict__ OUT)
{
    extern __shared__ float smem[];
    float* sg   = smem;              // g[t], t in [0, Teff); sg[ZSLOT] == 0
    float* sw   = smem + SW_OFF;     // w[k*9 + j]
    float* sAcc = smem + SACC_OFF;   // [0..7]=count_max sums, [8..15]=count_min

    const int tid  = threadIdx.x;
    const int lane = tid & 31;
    const int wav  = tid >> 5;
    const int nthr = blockDim.x;
    const int nwav = nthr >> 5;

    // block -> (b, u = di*2+diff_i, h)
    const int bid = blockIdx.x;
    const int b   = bid / (NUNIT * NGROUP);
    const int rem = bid % (NUNIT * NGROUP);
    const int u   = rem / NGROUP;
    const int h   = rem % NGROUP;
    const int di  = u >> 1;
    const int dif = u & 1;
    const int d   = 1 << di;
    const int Teff = dif ? (T_FULL - 1) : T_FULL;

    // ---- phase 0: init accumulators + zero slot, stage weights, channel idx ----
    if (tid < 16) sAcc[tid] = 0.0f;
    if (tid == 16) sg[ZSLOT] = 0.0f;
    {
        const int wbase = (u * (NKER * NGROUP) + h * NKER) * NTAP; // 72 consecutive floats
        if (tid < NKER * NTAP) sw[tid] = W[wbase + tid];
    }
    int ch[8];
    {
        const int* ip = IDX + (u * NGROUP + h) * 8;
        #pragma unroll
        for (int i = 0; i < 8; ++i) ch[i] = ip[i];
    }

    // ---- phase 1: materialize g (channel gather-sum, optional first diff) in LDS ----
    {
        const float* Xb = X + (size_t)b * CH_IN * T_FULL;
        for (int t = tid; t < Teff; t += nthr) {
            float s = 0.0f;
            #pragma unroll
            for (int i = 0; i < 8; ++i) {
                const float* xc = Xb + (size_t)ch[i] * T_FULL + t;
                float v = xc[0];
                if (dif) v = xc[1] - v;   // diff_X[t] = X[t+1] - X[t]
                s += v;
            }
            sg[t] = s;
        }
    }
    __syncthreads();

    // ---- WMMA A fragments: A[M=k (pad 16), K=j (9 pad 12)], 3 K-tiles of 4 ----
    const int c16 = lane & 15;            // M for A, N (time col) for B/D
    const int krb = (lane >> 4) << 1;     // lanes 0-15 -> K={0,1}; 16-31 -> K={2,3}
    v2f afrag[3];
    #pragma unroll
    for (int tile = 0; tile < 3; ++tile) {
        #pragma unroll
        for (int i = 0; i < 2; ++i) {
            const int j = tile * 4 + krb + i;
            afrag[tile][i] = (c16 < NKER && j < NTAP) ? sw[c16 * NTAP + j] : 0.0f;
        }
    }

    float accMax[8], accMin[8];
    #pragma unroll
    for (int k = 0; k < 8; ++k) { accMax[k] = 0.0f; accMin[k] = 0.0f; }

    // ---- phase 2: 32 time points / iter: 12 branchless ds_loads, then 6 WMMAs ----
    const int nblk = (Teff + 31) >> 5;
    for (int blk = wav; blk < nblk; blk += nwav) {
        const int t0 = blk << 5;

        // gather B fragments for both 16-wide half-blocks (address-select zero pad)
        v2f bf0[3], bf1[3];
        #pragma unroll
        for (int tile = 0; tile < 3; ++tile) {
            #pragma unroll
            for (int i = 0; i < 2; ++i) {
                const int j  = tile * 4 + krb + i;
                const int p0 = t0 + c16 + (j - 4) * d;       // half-block 0 tap
                const int p1 = p0 + 16;                      // half-block 1 tap
                const int a0 = (j < NTAP && (unsigned)p0 < (unsigned)Teff) ? p0 : ZSLOT;
                const int a1 = (j < NTAP && (unsigned)p1 < (unsigned)Teff) ? p1 : ZSLOT;
                bf0[tile][i] = sg[a0];
                bf1[tile][i] = sg[a1];
            }
        }

        v8f c0 = {0.f,0.f,0.f,0.f,0.f,0.f,0.f,0.f};
        v8f c1 = {0.f,0.f,0.f,0.f,0.f,0.f,0.f,0.f};
        #pragma unroll
        for (int tile = 0; tile < 3; ++tile) {   // interleave two accumulators
            c0 = __builtin_amdgcn_wmma_f32_16x16x4_f32(
                     false, afrag[tile], false, bf0[tile], (short)0, c0, false, false);
            c1 = __builtin_amdgcn_wmma_f32_16x16x4_f32(
                     false, afrag[tile], false, bf1[tile], (short)0, c1, false, false);
        }

        // pack: lanes 0-15 <- D0 (t0+lane), lanes 16-31 <- D1 lane-(lane-16) (t0+lane)
        float z[8];
        #pragma unroll
        for (int k = 0; k < 8; ++k) {
            const float hi = __shfl(c1[k], c16, 32);   // ds_bpermute
            z[k] = (lane < 16) ? c0[k] : hi;
        }

        if (t0 + lane < Teff) {
            float mx = z[0], mn = z[0];
            #pragma unroll
            for (int k = 1; k < 8; ++k) { mx = fmaxf(mx, z[k]); mn = fminf(mn, z[k]); }
            bool dmx = false, dmn = false;   // first-occurrence argmax/argmin semantics
            #pragma unroll
            for (int k = 0; k < 8; ++k) {
                const bool hm = (!dmx) && (z[k] == mx);
                accMax[k] += hm ? mx : 0.0f;  dmx = dmx || hm;
                const bool hn = (!dmn) && (z[k] == mn);
                accMin[k] += hn ? 1.0f : 0.0f; dmn = dmn || hn;
            }
        }
    }

    // ---- phase 3: wave butterfly reduce, then ds_add_f32 combine across waves ----
    #pragma unroll
    for (int k = 0; k < 8; ++k) {
        float vM = accMax[k], vN = accMin[k];
        #pragma unroll
        for (int off = 16; off > 0; off >>= 1) {
            vM += __shfl_xor(vM, off, 32);
            vN += __shfl_xor(vN, off, 32);
        }
        if (lane == 0) {
            atomicAdd(&sAcc[k],     vM);
            atomicAdd(&sAcc[8 + k], vN);
        }
    }
    __syncthreads();

    // ---- phase 4: this WG exclusively owns its 16 outputs -> plain stores ----
    if (tid < 16) {
        const int which = tid >> 3;   // 0 = count_max, 1 = count_min
        const int k     = tid & 7;
        const size_t off = (size_t)b * 14336
                         + (size_t)(2 * u + which) * (NGROUP * NKER)
                         + (size_t)h * NKER + k;
        OUT[off] = sAcc[tid];
    }
}

extern "C" void kernel_launch(void* const* d_in, const int* in_sizes, int n_in,
                              void* d_out, int out_size, void* d_ws, size_t ws_size,
                              hipStream_t stream) {
    (void)in_sizes; (void)n_in; (void)d_ws; (void)ws_size; (void)out_size;
    const float* X   = (const float*)d_in[0];
    const float* W   = (const float*)d_in[1];
    const int*   IDX = (const int*)d_in[2];
    float*       OUT = (float*)d_out;

    const int shmem = SMEM_FLOATS * (int)sizeof(float);   // ~293.5 KB, fits 320 KB WGP LDS
    (void)hipFuncSetAttribute((const void*)hydra_unit_kernel,
                              hipFuncAttributeMaxDynamicSharedMemorySize, shmem);

    const int nblocks = 4 * NUNIT * NGROUP;   // 3584 independent units
    hydra_unit_kernel<<<dim3(nblocks), dim3(1024), shmem, stream>>>(X, W, IDX, OUT);
}